// CustomMLPLayer_10806137716759
// MI455X (gfx1250) — compile-verified
//
#include <hip/hip_runtime.h>
#include <hip/hip_bf16.h>

// Problem sizes (from the reference)
#define D_MODEL 4096
#define D_FF    11008
#define SEQ     2048

typedef __attribute__((ext_vector_type(2))) float v2f;
typedef __attribute__((ext_vector_type(8))) float v8f;
typedef __attribute__((ext_vector_type(4))) int   v4i;

// GEMM tiling: C[M=SEQ][N=D_MODEL] = x[M][K] * W[N][K]^T, K = D_FF
#define BM 128
#define BN 128
#define BK 16
#define LDS_STRIDE 20   // 20 floats = 80B row stride: 16B aligned, bank-conflict free

// gfx1250 async memory->LDS path (ASYNCcnt); builtin confirmed to exist by the
// round-2 diagnostic: param 0 is `int4 addrspace(1)*` (global), param 1 LDS.
#if defined(__gfx1250__) && \
    __has_builtin(__builtin_amdgcn_global_load_async_to_lds_b128) && \
    __has_builtin(__builtin_amdgcn_s_wait_asynccnt)
#define USE_ASYNC_LDS 1
typedef __attribute__((address_space(1))) v4i* gvp_t;  // global int4*
typedef __attribute__((address_space(3))) v4i* lvp_t;  // LDS int4*
#else
#define USE_ASYNC_LDS 0
#endif

__device__ __forceinline__ void issue_tile_loads(const float* __restrict__ x,
                                                 const float* __restrict__ W,
                                                 float* as, float* bs,
                                                 int bm, int bn, int kt, int tid) {
  // Stage one 128x16 tile of each operand: 512 x b128 per matrix, 2 per thread.
#pragma unroll
  for (int i = 0; i < 2; ++i) {
    const int idx = tid + i * 256;
    const int row = idx >> 2;   // 0..127
    const int c4  = idx & 3;    // which float4 of the 16-wide row
    const float* ga = x + (size_t)(bm + row) * D_FF + kt + c4 * 4;
    const float* gb = W + (size_t)(bn + row) * D_FF + kt + c4 * 4;
    float* la = &as[row * LDS_STRIDE + c4 * 4];
    float* lb = &bs[row * LDS_STRIDE + c4 * 4];
#if USE_ASYNC_LDS
    // Direct memory->LDS DMA, no VGPR staging (GLOBAL_LOAD_ASYNC_TO_LDS_B128)
    __builtin_amdgcn_global_load_async_to_lds_b128((gvp_t)ga, (lvp_t)la, 0, 0);
    __builtin_amdgcn_global_load_async_to_lds_b128((gvp_t)gb, (lvp_t)lb, 0, 0);
#else
    *reinterpret_cast<float4*>(la) = *reinterpret_cast<const float4*>(ga);
    *reinterpret_cast<float4*>(lb) = *reinterpret_cast<const float4*>(gb);
#endif
  }
}

__global__ __launch_bounds__(256)
void gemm_wmma_f32_kernel(const float* __restrict__ x,
                          const float* __restrict__ W,
                          float* __restrict__ out) {
  __shared__ float As[2][BM * LDS_STRIDE];  // double-buffered x tile  [BM][BK]
  __shared__ float Bs[2][BN * LDS_STRIDE];  // double-buffered W tile  [BN][BK]

  const int tid    = threadIdx.x;
  const int wid    = tid >> 5;     // 0..7 (wave32)
  const int lane   = tid & 31;
  const int wave_m = wid >> 1;     // 0..3 -> 32 rows each
  const int wave_n = wid & 1;      // 0..1 -> 64 cols each
  const int lrow   = lane & 15;    // fragment row / col index
  const int lkh    = lane >> 4;    // K half select: lanes 16-31 hold K+2..K+3

  const int bm = blockIdx.y * BM;
  const int bn = blockIdx.x * BN;

  v8f acc[2][4] = {};  // 2 (M frags) x 4 (N frags) of 16x16 f32

  // Prologue: start loads for tile 0 into buffer 0
  issue_tile_loads(x, W, &As[0][0], &Bs[0][0], bm, bn, 0, tid);

  int buf = 0;
  for (int kt = 0; kt < D_FF; kt += BK) {
#if USE_ASYNC_LDS
    __builtin_amdgcn_s_wait_asynccnt(0);   // my tile-`buf` DMAs have landed
#endif
    __syncthreads();                        // all waves' data visible; prev compute done

    // Overlap: kick off DMAs for the next K-tile into the other buffer
    if (kt + BK < D_FF)
      issue_tile_loads(x, W, &As[buf ^ 1][0], &Bs[buf ^ 1][0], bm, bn, kt + BK, tid);

    const float* as = &As[buf][0];
    const float* bs = &Bs[buf][0];

    // ---- 4 WMMA k-steps of K=4 each: 32 v_wmma per tile per wave ----
#pragma unroll
    for (int kk = 0; kk < BK; kk += 4) {
      v2f afrag[2], bfrag[4];
      // A fragment (16x4 f32, ISA 7.12.2): lane L holds row M=L%16,
      // VGPR v holds K = kk + (L/16)*2 + v
#pragma unroll
      for (int mi = 0; mi < 2; ++mi) {
        const float* ap =
            &as[(wave_m * 32 + mi * 16 + lrow) * LDS_STRIDE + kk + lkh * 2];
        afrag[mi].x = ap[0];
        afrag[mi].y = ap[1];
      }
      // B fragment (4x16): W is N-major/K-contiguous, mirrored layout
#pragma unroll
      for (int ni = 0; ni < 4; ++ni) {
        const float* bp =
            &bs[(wave_n * 64 + ni * 16 + lrow) * LDS_STRIDE + kk + lkh * 2];
        bfrag[ni].x = bp[0];
        bfrag[ni].y = bp[1];
      }
#pragma unroll
      for (int mi = 0; mi < 2; ++mi)
#pragma unroll
        for (int ni = 0; ni < 4; ++ni)
          acc[mi][ni] = __builtin_amdgcn_wmma_f32_16x16x4_f32(
              /*neg_a=*/false, afrag[mi], /*neg_b=*/false, bfrag[ni],
              /*c_mod=*/(short)0, acc[mi][ni],
              /*reuse_a=*/false, /*reuse_b=*/false);
    }
    buf ^= 1;
    // No trailing barrier: next iteration's wait+barrier protects buffer reuse.
  }

  // ---- epilogue: C/D layout: lane L<16 -> N=L,M=v ; L>=16 -> N=L-16,M=v+8
#pragma unroll
  for (int mi = 0; mi < 2; ++mi) {
#pragma unroll
    for (int ni = 0; ni < 4; ++ni) {
      const int col = bn + wave_n * 64 + ni * 16 + lrow;
#pragma unroll
      for (int v = 0; v < 8; ++v) {
        const int row = bm + wave_m * 32 + mi * 16 + v + lkh * 8;
        out[(size_t)row * D_MODEL + col] = acc[mi][ni][v];
      }
    }
  }
}

// counts[f] = #tokens with x[s][f] > 0 (ReLU activation count per neuron)
__global__ __launch_bounds__(256)
void neuron_count_kernel(const float* __restrict__ x, float* __restrict__ counts) {
  const int f = blockIdx.x * blockDim.x + threadIdx.x;
  if (f >= D_FF) return;
  int c = 0;
  for (int s = 0; s < SEQ; ++s) c += (x[(size_t)s * D_FF + f] > 0.0f) ? 1 : 0;
  counts[f] = (float)c;
}

// single block: mean over counts, cutoff = floor(mean * 0.95), mask = counts > cutoff
__global__ __launch_bounds__(256)
void neuron_mask_kernel(const float* __restrict__ counts, float* __restrict__ mask_out) {
  __shared__ float red[256];
  float sum = 0.0f;
  for (int f = threadIdx.x; f < D_FF; f += 256) sum += counts[f];
  red[threadIdx.x] = sum;
  __syncthreads();
  for (int s = 128; s > 0; s >>= 1) {
    if (threadIdx.x < s) red[threadIdx.x] += red[threadIdx.x + s];
    __syncthreads();
  }
  const float cutoff = floorf((red[0] / (float)D_FF) * 0.95f);
  for (int f = threadIdx.x; f < D_FF; f += 256)
    mask_out[f] = (counts[f] > cutoff) ? 1.0f : 0.0f;
}

extern "C" void kernel_launch(void* const* d_in, const int* in_sizes, int n_in,
                              void* d_out, int out_size, void* d_ws, size_t ws_size,
                              hipStream_t stream) {
  (void)in_sizes; (void)n_in; (void)out_size; (void)ws_size;
  const float* x = (const float*)d_in[0];   // [1, SEQ, D_FF] f32
  const float* W = (const float*)d_in[1];   // [D_MODEL, D_FF] f32
  float* out      = (float*)d_out;                         // [SEQ, D_MODEL]
  float* mask_out = (float*)d_out + (size_t)SEQ * D_MODEL; // [D_FF]
  float* counts   = (float*)d_ws;                          // [D_FF] scratch

  dim3 grid(D_MODEL / BN, SEQ / BM);  // 32 x 16 workgroups
  gemm_wmma_f32_kernel<<<grid, 256, 0, stream>>>(x, W, out);

  neuron_count_kernel<<<(D_FF + 255) / 256, 256, 0, stream>>>(x, counts);
  neuron_mask_kernel<<<1, 256, 0, stream>>>(counts, mask_out);
}